// MambaModel_14379550507699
// MI455X (gfx1250) — compile-verified
//
#include <hip/hip_runtime.h>
#include <hip/hip_bf16.h>

#define VOCAB   50257
#define D_MODEL 512
#define N_LAYERS 4
#define D_STATE 64
#define D_CONV  4
#define D_INNER 1024
#define HEADDIM 64
#define NHEADS  16
#define D_IN_PROJ (2*D_INNER + 2*D_STATE + NHEADS)  // 2192
#define CONV_DIM  (D_INNER + 2*D_STATE)             // 1152
#define BATCH 2
#define SEQ   1024
#define ROWS  (BATCH*SEQ)                           // 2048
#define EPS   1e-5f

typedef float v2f __attribute__((ext_vector_type(2)));
typedef float v8f __attribute__((ext_vector_type(8)));

// ---------------------------------------------------------------------------
// Embedding gather: x[row,:] = embed_w[tokens[row],:]
// ---------------------------------------------------------------------------
__global__ __launch_bounds__(256) void embed_kernel(const int* __restrict__ tokens,
                                                    const float* __restrict__ embed_w,
                                                    float* __restrict__ x) {
    const int row = blockIdx.x;
    const int tok = tokens[row];
    const float* src = embed_w + (size_t)tok * D_MODEL;
    float* dst = x + (size_t)row * D_MODEL;
    for (int i = threadIdx.x; i < D_MODEL; i += blockDim.x) dst[i] = src[i];
}

// ---------------------------------------------------------------------------
// WMMA fp32 GEMM:  C[M,N] = A[M,K] @ W[N,K]^T (+ bias)
// One wave -> 16(M) x 64(N) of C. Per k-step the A fragment is reused across
// 4 V_WMMA_F32_16X16X4_F32 issues (5 float2 loads -> 4 WMMAs), and the k-loop
// is software-pipelined 2 deep: loads for step kk+2 are issued before the
// WMMAs consuming step kk, so s_wait_loadcnt overlaps VMEM with the matrix
// pipe instead of draining to zero each step.
// Fragment layouts (ISA 7.12.2, fp32 16x16x4):
//   A (16x4 MxK): lane l<16 -> M=l; v0 = K=(0|2), v1 = K=(1|3) by lane-half.
//   B (4x16 KxN): identical load pattern on W rows (N role), W stored [N,K].
//   C/D: lane%16 = N, vgpr r -> M = r + 8*(lane/16).
// Requires M%16==0, K%16==0 (KT even); N edge: clamped loads, guarded stores.
// WMMAs stay in wave-uniform control flow (EXEC all-1s requirement).
// ---------------------------------------------------------------------------
__global__ __launch_bounds__(256) void wmma_gemm_kernel(const float* __restrict__ A, int lda,
                                                        const float* __restrict__ W, int ldw,
                                                        const float* __restrict__ bias,
                                                        float* __restrict__ C, int ldc,
                                                        int M, int N, int K) {
    const int lane = threadIdx.x & 31;
    const int wave = threadIdx.x >> 5;
    const int half = lane >> 4;      // 0 or 1
    const int lmod = lane & 15;

    const int nTiles64 = (N + 63) >> 6;
    const int tile64 = blockIdx.x * 8 + wave;    // 8 waves/block span N
    if (tile64 >= nTiles64) return;              // wave-uniform exit
    const int tileM = blockIdx.y;

    const int rowA = tileM * 16 + lmod;
    const float* aptr = A + (size_t)rowA * lda + 2 * half;

    const float* wptr[4];
    int col[4];
#pragma unroll
    for (int s = 0; s < 4; ++s) {
        col[s] = tile64 * 64 + s * 16 + lmod;
        const int rw = (col[s] < N) ? col[s] : (N - 1);   // clamp; store guarded
        wptr[s] = W + (size_t)rw * ldw + 2 * half;
    }

    v8f acc0 = {}, acc1 = {}, acc2 = {}, acc3 = {};

    // 2-stage pipeline buffers (stage = one k-step of 4)
    v2f abuf[2];
    v2f bbuf[2][4];
#pragma unroll
    for (int s = 0; s < 2; ++s) {
        abuf[s] = *(const v2f*)(aptr + 4 * s);
#pragma unroll
        for (int j = 0; j < 4; ++j)
            bbuf[s][j] = *(const v2f*)(wptr[j] + 4 * s);
    }

    const int KT = K >> 2;            // k-steps; even for all GEMMs here
#pragma unroll 2
    for (int kk = 0; kk < KT - 2; ++kk) {
        const int cur = kk & 1;
        const int pk = (kk + 2) << 2;
        // issue next-next stage loads first (independent of this step's WMMAs)
        v2f an = *(const v2f*)(aptr + pk);
        v2f bn0 = *(const v2f*)(wptr[0] + pk);
        v2f bn1 = *(const v2f*)(wptr[1] + pk);
        v2f bn2 = *(const v2f*)(wptr[2] + pk);
        v2f bn3 = *(const v2f*)(wptr[3] + pk);

        acc0 = __builtin_amdgcn_wmma_f32_16x16x4_f32(false, abuf[cur], false, bbuf[cur][0], (short)0, acc0, false, false);
        acc1 = __builtin_amdgcn_wmma_f32_16x16x4_f32(false, abuf[cur], false, bbuf[cur][1], (short)0, acc1, false, false);
        acc2 = __builtin_amdgcn_wmma_f32_16x16x4_f32(false, abuf[cur], false, bbuf[cur][2], (short)0, acc2, false, false);
        acc3 = __builtin_amdgcn_wmma_f32_16x16x4_f32(false, abuf[cur], false, bbuf[cur][3], (short)0, acc3, false, false);

        abuf[cur] = an;
        bbuf[cur][0] = bn0; bbuf[cur][1] = bn1;
        bbuf[cur][2] = bn2; bbuf[cur][3] = bn3;
    }
    // epilogue: drain the last two stages
#pragma unroll
    for (int e = 0; e < 2; ++e) {
        const int cur = (KT - 2 + e) & 1;
        acc0 = __builtin_amdgcn_wmma_f32_16x16x4_f32(false, abuf[cur], false, bbuf[cur][0], (short)0, acc0, false, false);
        acc1 = __builtin_amdgcn_wmma_f32_16x16x4_f32(false, abuf[cur], false, bbuf[cur][1], (short)0, acc1, false, false);
        acc2 = __builtin_amdgcn_wmma_f32_16x16x4_f32(false, abuf[cur], false, bbuf[cur][2], (short)0, acc2, false, false);
        acc3 = __builtin_amdgcn_wmma_f32_16x16x4_f32(false, abuf[cur], false, bbuf[cur][3], (short)0, acc3, false, false);
    }

    v8f accs[4] = {acc0, acc1, acc2, acc3};
#pragma unroll
    for (int s = 0; s < 4; ++s) {
        if (col[s] < N) {
            const float bv = bias ? bias[col[s]] : 0.0f;
#pragma unroll
            for (int r = 0; r < 8; ++r) {
                const int m = tileM * 16 + r + 8 * half;
                C[(size_t)m * ldc + col[s]] = accs[s][r] + bv;
            }
        }
    }
}

// ---------------------------------------------------------------------------
// Depthwise causal conv (k=4) over xBC slice of zxbcdt, + bias, + SiLU.
// Also computes dt = softplus(zxbcdt[..., -16:] + dt_bias).
// One block per (b,l) row.
// ---------------------------------------------------------------------------
__device__ __forceinline__ float silu_f(float v) { return v / (1.0f + __expf(-v)); }

__global__ __launch_bounds__(256) void conv_silu_kernel(const float* __restrict__ zxbcdt,
                                                        const float* __restrict__ conv_w,
                                                        const float* __restrict__ conv_b,
                                                        const float* __restrict__ dt_bias,
                                                        float* __restrict__ xBCc,
                                                        float* __restrict__ dt) {
    const int row = blockIdx.x;          // b*SEQ + l
    const int l   = row & (SEQ - 1);
    const float* base = zxbcdt + (size_t)row * D_IN_PROJ + D_INNER;  // xBC slice
    for (int c = threadIdx.x; c < CONV_DIM; c += blockDim.x) {
        float s = conv_b[c];
#pragma unroll
        for (int k = 0; k < D_CONV; ++k) {
            const int dl = k - (D_CONV - 1);     // -3..0
            if (l + dl >= 0)
                s += base[(long long)dl * D_IN_PROJ + c] * conv_w[c * D_CONV + k];
        }
        xBCc[(size_t)row * CONV_DIM + c] = silu_f(s);
    }
    if (threadIdx.x < NHEADS) {
        const int h = threadIdx.x;
        float v = zxbcdt[(size_t)row * D_IN_PROJ + (D_IN_PROJ - NHEADS) + h] + dt_bias[h];
        float sp = (v > 20.0f) ? v : log1pf(__expf(v));
        dt[(size_t)row * NHEADS + h] = sp;
    }
}

// ---------------------------------------------------------------------------
// Sequential SSD scan. One block per (b,h); 256 threads own the 64x64 state:
// thread t -> p = t/4, n in [ (t%4)*16, +16 ).  B_t/C_t staged through LDS.
// y[b,l,h,p] = sum_n h[p,n]*C[n] + D[h]*x, with h = exp(dt*A)*h + dt*x*B.
// ---------------------------------------------------------------------------
__global__ __launch_bounds__(256) void scan_kernel(const float* __restrict__ xBCc,
                                                   const float* __restrict__ dt,
                                                   const float* __restrict__ A_log,
                                                   const float* __restrict__ Dvec,
                                                   float* __restrict__ y) {
    const int bh = blockIdx.x;
    const int b = bh / NHEADS, h = bh % NHEADS;
    const int tid = threadIdx.x;
    const int p   = tid >> 2;        // 0..63
    const int nq  = tid & 3;
    const int n0  = nq * 16;

    const float A  = -__expf(A_log[h]);
    const float Dh = Dvec[h];

    float hs[16];
#pragma unroll
    for (int j = 0; j < 16; ++j) hs[j] = 0.0f;

    __shared__ float Bs[D_STATE];
    __shared__ float Cs[D_STATE];

    for (int t = 0; t < SEQ; ++t) {
        const size_t row = (size_t)b * SEQ + t;
        if (tid < 2 * D_STATE) {
            const float v = xBCc[row * CONV_DIM + D_INNER + tid];
            if (tid < D_STATE) Bs[tid] = v; else Cs[tid - D_STATE] = v;
        }
        __syncthreads();

        const float dtv = dt[row * NHEADS + h];
        const float dA  = __expf(dtv * A);
        const float xv  = xBCc[row * CONV_DIM + h * HEADDIM + p];
        const float cf  = dtv * xv;

        float part = 0.0f;
#pragma unroll
        for (int j = 0; j < 16; ++j) {
            hs[j] = dA * hs[j] + cf * Bs[n0 + j];
            part += hs[j] * Cs[n0 + j];
        }
        part += __shfl_xor(part, 1, 32);
        part += __shfl_xor(part, 2, 32);
        if (nq == 0)
            y[row * D_INNER + h * HEADDIM + p] = part + Dh * xv;
        __syncthreads();
    }
}

// ---------------------------------------------------------------------------
// g = y * silu(z);  g *= rsqrt(mean(g^2)+eps) * norm_w.  One block per row.
// ---------------------------------------------------------------------------
__global__ __launch_bounds__(256) void gate_norm_kernel(const float* __restrict__ y,
                                                        const float* __restrict__ zxbcdt,
                                                        const float* __restrict__ norm_w,
                                                        float* __restrict__ gn) {
    const int row = blockIdx.x;
    const float* z  = zxbcdt + (size_t)row * D_IN_PROJ;  // z = first D_INNER cols
    const float* yr = y + (size_t)row * D_INNER;

    float g[4];
    float ss = 0.0f;
#pragma unroll
    for (int i = 0; i < 4; ++i) {
        const int c = threadIdx.x + i * 256;
        const float gv = yr[c] * silu_f(z[c]);
        g[i] = gv;
        ss += gv * gv;
    }
#pragma unroll
    for (int off = 16; off > 0; off >>= 1) ss += __shfl_xor(ss, off, 32);
    __shared__ float red[8];
    if ((threadIdx.x & 31) == 0) red[threadIdx.x >> 5] = ss;
    __syncthreads();
    float tot = 0.0f;
#pragma unroll
    for (int i = 0; i < 8; ++i) tot += red[i];
    const float sc = rsqrtf(tot * (1.0f / D_INNER) + EPS);
#pragma unroll
    for (int i = 0; i < 4; ++i) {
        const int c = threadIdx.x + i * 256;
        gn[(size_t)row * D_INNER + c] = g[i] * sc * norm_w[c];
    }
}

// ---------------------------------------------------------------------------
// Final LayerNorm over D_MODEL=512. One block (256 thr, 2 elems/thr) per row.
// ---------------------------------------------------------------------------
__global__ __launch_bounds__(256) void ln_kernel(const float* __restrict__ x,
                                                 const float* __restrict__ w,
                                                 const float* __restrict__ bsh,
                                                 float* __restrict__ out) {
    const int row = blockIdx.x;
    const float* xr = x + (size_t)row * D_MODEL;
    const int c0 = threadIdx.x, c1 = threadIdx.x + 256;
    const float v0 = xr[c0], v1 = xr[c1];
    float s  = v0 + v1;
    float sq = v0 * v0 + v1 * v1;
#pragma unroll
    for (int off = 16; off > 0; off >>= 1) {
        s  += __shfl_xor(s,  off, 32);
        sq += __shfl_xor(sq, off, 32);
    }
    __shared__ float rs[8], rq[8];
    if ((threadIdx.x & 31) == 0) { rs[threadIdx.x >> 5] = s; rq[threadIdx.x >> 5] = sq; }
    __syncthreads();
    float ts = 0.0f, tq = 0.0f;
#pragma unroll
    for (int i = 0; i < 8; ++i) { ts += rs[i]; tq += rq[i]; }
    const float mu  = ts * (1.0f / D_MODEL);
    const float var = tq * (1.0f / D_MODEL) - mu * mu;
    const float inv = rsqrtf(var + EPS);
    out[(size_t)row * D_MODEL + c0] = (v0 - mu) * inv * w[c0] + bsh[c0];
    out[(size_t)row * D_MODEL + c1] = (v1 - mu) * inv * w[c1] + bsh[c1];
}

// ---------------------------------------------------------------------------
extern "C" void kernel_launch(void* const* d_in, const int* in_sizes, int n_in,
                              void* d_out, int out_size, void* d_ws, size_t ws_size,
                              hipStream_t stream) {
    (void)in_sizes; (void)n_in; (void)out_size; (void)ws_size;
    const int*   tokens  = (const int*)  d_in[0];
    const float* embed_w = (const float*)d_in[1];
    const float* in_w    = (const float*)d_in[2];   // (4, 2192, 512)
    const float* conv_w  = (const float*)d_in[3];   // (4, 1152, 4)
    const float* conv_b  = (const float*)d_in[4];   // (4, 1152)
    const float* dt_bias = (const float*)d_in[5];   // (4, 16)
    const float* A_log   = (const float*)d_in[6];   // (4, 16)
    const float* Dvec    = (const float*)d_in[7];   // (4, 16)
    const float* norm_w  = (const float*)d_in[8];   // (4, 1024)
    const float* out_w   = (const float*)d_in[9];   // (4, 512, 1024)
    const float* ln_w    = (const float*)d_in[10];  // (512,)
    const float* ln_b    = (const float*)d_in[11];  // (512,)
    const float* head_w  = (const float*)d_in[12];  // (50257, 512)
    const float* head_b  = (const float*)d_in[13];  // (50257,)
    float* out = (float*)d_out;                     // (2, 1024, 50257) fp32

    // Workspace layout (~48.5 MB of fp32)
    float* x    = (float*)d_ws;                     // ROWS * D_MODEL
    float* zx   = x    + (size_t)ROWS * D_MODEL;    // ROWS * D_IN_PROJ
    float* xBCc = zx   + (size_t)ROWS * D_IN_PROJ;  // ROWS * CONV_DIM
    float* dtb  = xBCc + (size_t)ROWS * CONV_DIM;   // ROWS * NHEADS
    float* yb   = dtb  + (size_t)ROWS * NHEADS;     // ROWS * D_INNER
    float* gn   = yb   + (size_t)ROWS * D_INNER;    // ROWS * D_INNER (also reused as xln)

    embed_kernel<<<ROWS, 256, 0, stream>>>(tokens, embed_w, x);

    for (int l = 0; l < N_LAYERS; ++l) {
        const float* inw = in_w  + (size_t)l * D_IN_PROJ * D_MODEL;
        const float* ow  = out_w + (size_t)l * D_MODEL * D_INNER;

        // in_proj: zx = x @ in_w^T   (M=2048, N=2192, K=512)
        {
            const int nt64 = (D_IN_PROJ + 63) / 64;
            dim3 grid((nt64 + 7) / 8, ROWS / 16);
            wmma_gemm_kernel<<<grid, 256, 0, stream>>>(x, D_MODEL, inw, D_MODEL,
                                                       nullptr, zx, D_IN_PROJ,
                                                       ROWS, D_IN_PROJ, D_MODEL);
        }

        conv_silu_kernel<<<ROWS, 256, 0, stream>>>(zx,
                                                   conv_w + (size_t)l * CONV_DIM * D_CONV,
                                                   conv_b + (size_t)l * CONV_DIM,
                                                   dt_bias + (size_t)l * NHEADS,
                                                   xBCc, dtb);

        scan_kernel<<<BATCH * NHEADS, 256, 0, stream>>>(xBCc, dtb,
                                                        A_log + (size_t)l * NHEADS,
                                                        Dvec  + (size_t)l * NHEADS, yb);

        gate_norm_kernel<<<ROWS, 256, 0, stream>>>(yb, zx,
                                                   norm_w + (size_t)l * D_INNER, gn);

        // out_proj: x = gn @ out_w^T  (M=2048, N=512, K=1024)
        {
            const int nt64 = (D_MODEL + 63) / 64;
            dim3 grid((nt64 + 7) / 8, ROWS / 16);
            wmma_gemm_kernel<<<grid, 256, 0, stream>>>(gn, D_INNER, ow, D_INNER,
                                                       nullptr, x, D_MODEL,
                                                       ROWS, D_MODEL, D_INNER);
        }
    }

    // Final LayerNorm (result into gn, reused as xln: ROWS x D_MODEL)
    ln_kernel<<<ROWS, 256, 0, stream>>>(x, ln_w, ln_b, gn);

    // LM head: out = xln @ head_w^T + head_b  (M=2048, N=50257, K=512)
    {
        const int nt64 = (VOCAB + 63) / 64;   // 786 (edge tile guarded)
        dim3 grid((nt64 + 7) / 8, ROWS / 16);
        wmma_gemm_kernel<<<grid, 256, 0, stream>>>(gn, D_MODEL, head_w, D_MODEL,
                                                   head_b, out, VOCAB,
                                                   ROWS, VOCAB, D_MODEL);
    }
}